// FieldFactorizationMachine_25580825215408
// MI455X (gfx1250) — compile-verified
//
#include <hip/hip_runtime.h>

#define FFM_F 32
#define FFM_L 16
#define FFM_ROW (FFM_F * FFM_L)   // 512 floats per embedding row
#define LDS_STRIDE 520            // padded row stride (floats) to reduce bank conflicts
#define THREADS 256

typedef __attribute__((ext_vector_type(2))) float v2f;
typedef __attribute__((ext_vector_type(4))) float v4f;
typedef __attribute__((ext_vector_type(8))) float v8f;
typedef __attribute__((ext_vector_type(4))) int   v4i;

typedef __attribute__((address_space(1))) v4i* gv4i_p;   // global int4*
typedef __attribute__((address_space(3))) v4i* lv4i_p;   // LDS int4*

__global__ __launch_bounds__(THREADS) void ffm_kernel(
    const int* __restrict__ x,            // [B, 32]
    const float* __restrict__ weights,    // [FEAT, 32, 16]
    const float* __restrict__ bias_w,     // [FEAT, 1]
    float* __restrict__ out)              // [B]
{
    extern __shared__ float sm[];         // 32 rows * LDS_STRIDE floats (+ reuse for reduction)
    const int b = blockIdx.x;
    const int t = threadIdx.x;
    const int i = t >> 3;                 // field row 0..31 (8 threads per row)
    const int u = t & 7;                  // sub-lane within row

    const int idx = x[b * FFM_F + i];     // uniform per 8 threads (scalar-cached)
    const float* grow = weights + (size_t)idx * FFM_ROW;
    float* lrow = sm + i * LDS_STRIDE;

    // ---- Stage the 32 gathered rows (64KB) into LDS via async global->LDS DMA ----
#if __has_builtin(__builtin_amdgcn_global_load_async_to_lds_b128)
    #pragma unroll
    for (int s = 0; s < 16; ++s) {
        const int off = u * 4 + s * 32;   // 16B per lane, 128B contiguous per row per wave-op
        __builtin_amdgcn_global_load_async_to_lds_b128(
            (gv4i_p)(void*)(grow + off), (lv4i_p)(void*)(lrow + off), 0, 0);
    }
  #if __has_builtin(__builtin_amdgcn_s_wait_asynccnt)
    __builtin_amdgcn_s_wait_asynccnt(0);
  #else
    asm volatile("s_wait_asynccnt 0" ::: "memory");
  #endif
#else
    #pragma unroll
    for (int s = 0; s < 16; ++s) {
        const int off = u * 4 + s * 32;
        *(v4f*)(lrow + off) = *(const v4f*)(grow + off);
    }
#endif
    __syncthreads();

    // ---- S_full partial products: thread covers row i, k = u*4 + 32*s (+0..3) ----
    // term(i,k) = M[i][k] * M[k>>4][ i*16 + (k&15) ]
    float P = 0.0f;
    const int poff = i * FFM_L + (u & 3) * 4;  // partner inner offset (k&15 == (u&3)*4)
    const int jbase = (u >> 2);                // (k>>4) == (u>>2) + 2*s
    #pragma unroll
    for (int s = 0; s < 16; ++s) {
        const int k = u * 4 + s * 32;
        const v4f a = *(const v4f*)(lrow + k);
        const v4f p = *(const v4f*)(sm + (jbase + 2 * s) * LDS_STRIDE + poff);
        P += a.x * p.x + a.y * p.y + a.z * p.z + a.w * p.w;
    }

    float partial = 0.5f * P;
    if (t < FFM_F) {                      // wave-uniform branch (wave 0 only): EXEC stays full
        const float bsum = bias_w[x[b * FFM_F + t]];
        const float* rt = sm + t * LDS_STRIDE + t * FFM_L;
        float dsum = 0.0f;
        #pragma unroll
        for (int l = 0; l < FFM_L; ++l) dsum += rt[l] * rt[l];
        partial += bsum - 0.5f * dsum;    // bias and strict-triangle diagonal correction
    }

    // ---- Wave(32)-level sum of `partial` using V_WMMA_F32_16X16X4_F32 ----
    // A[m,:] = [p_m, 0, p_{m+16}, 0] (ISA 16x4 f32 A layout), B = ones(4x16)
    // => D[m,n] = p_m + p_{m+16}. Sum of a lane's 8 D-VGPRs covers rows m..m+7;
    // lane0 + lane16 covers all 16 rows => full 32-lane sum.
    float wsum;
#if __has_builtin(__builtin_amdgcn_wmma_f32_16x16x4_f32)
    {
        v2f a;  a.x = partial; a.y = 0.0f;
        v2f one; one.x = 1.0f; one.y = 1.0f;
        v8f c = {0.0f, 0.0f, 0.0f, 0.0f, 0.0f, 0.0f, 0.0f, 0.0f};
        v8f d = __builtin_amdgcn_wmma_f32_16x16x4_f32(
            false, a, false, one, (short)0, c, false, false);
        const float s8 = d[0] + d[1] + d[2] + d[3] + d[4] + d[5] + d[6] + d[7];
        wsum = __shfl(s8, 0, 32) + __shfl(s8, 16, 32);
    }
#else
    {
        float v = partial;
        #pragma unroll
        for (int off = 16; off > 0; off >>= 1) v += __shfl_down(v, off, 32);
        wsum = __shfl(v, 0, 32);
    }
#endif

    // ---- Cross-wave reduction (8 waves) via LDS, then store ----
    __syncthreads();                       // all LDS reads of sm[] complete before reuse
    if ((t & 31) == 0) sm[t >> 5] = wsum;
    __syncthreads();
    if (t == 0) {
        float tot = 0.0f;
        #pragma unroll
        for (int w = 0; w < THREADS / 32; ++w) tot += sm[w];
        out[b] = tot;
    }
}

extern "C" void kernel_launch(void* const* d_in, const int* in_sizes, int n_in,
                              void* d_out, int out_size, void* d_ws, size_t ws_size,
                              hipStream_t stream) {
    const int*   x  = (const int*)d_in[0];     // [4096, 32] int32
    const float* w  = (const float*)d_in[1];   // [200000, 32, 16] f32
    const float* bw = (const float*)d_in[2];   // [200000, 1] f32
    float* out = (float*)d_out;                // [4096] f32

    const int batch = in_sizes[0] / FFM_F;     // 4096
    const size_t shmem = (size_t)FFM_F * LDS_STRIDE * sizeof(float);  // 66560 B

    hipLaunchKernelGGL(ffm_kernel, dim3(batch), dim3(THREADS), shmem, stream,
                       x, w, bw, out);
}